// GCN_7395933684274
// MI455X (gfx1250) — compile-verified
//
#include <hip/hip_runtime.h>
#include <hip/hip_bf16.h>
#include <math.h>

typedef __attribute__((ext_vector_type(2))) float v2f;
typedef __attribute__((ext_vector_type(8))) float v8f;

#define HCH 128           // hidden channels
#define MAXV 119
#define NFEAT 9

// ---------------------------------------------------------------------------
// Atom encoder: h[n, c] = sum_f atom_emb[f, x[n,f], c]
// one block per node, 128 threads = channels
// ---------------------------------------------------------------------------
__global__ __launch_bounds__(HCH) void gcn_atom_encode(
    const int* __restrict__ x, const float* __restrict__ emb,
    float* __restrict__ h, int N) {
  int n = blockIdx.x;
  int c = threadIdx.x;
  __shared__ int xs[NFEAT];
  if (c < NFEAT) xs[c] = x[n * NFEAT + c];
  __syncthreads();
  float acc = 0.f;
#pragma unroll
  for (int f = 0; f < NFEAT; ++f)
    acc += emb[((size_t)f * MAXV + xs[f]) * HCH + c];
  h[(size_t)n * HCH + c] = acc;
}

// ---------------------------------------------------------------------------
// small helpers
// ---------------------------------------------------------------------------
__global__ void gcn_fill(float* __restrict__ p, float v, int n) {
  int i = blockIdx.x * blockDim.x + threadIdx.x;
  if (i < n) p[i] = v;
}

__global__ void gcn_count_deg(const int* __restrict__ ei, float* __restrict__ deg, int E) {
  int e = blockIdx.x * blockDim.x + threadIdx.x;
  if (e < E) atomicAdd(&deg[ei[E + e]], 1.0f);  // dst row of edge_index
}

__global__ void gcn_rsqrt_inplace(float* __restrict__ p, int n) {
  int i = blockIdx.x * blockDim.x + threadIdx.x;
  if (i < n) p[i] = rsqrtf(p[i]);   // deg >= 1 always (self loop)
}

// ---------------------------------------------------------------------------
// GEMM + fused self-loop epilogue using V_WMMA_F32_16X16X4_F32.
//   m[N,128]   = h[N,128] * W[128,128]
//   agg[N,128] = bias + m * dinv^2          (self-loop contribution)
// Block = 256 threads = 8 waves; block covers 32 rows x 128 cols.
// Wave w owns 16-col tile w and computes TWO stacked 16x16 M-tiles that
// share each B fragment (halves B load traffic, two independent WMMA
// chains overlap XDL latency). A tile (32x128 f32 = 16 KB) staged in LDS.
//
// f32 WMMA fragment layouts (CDNA5 ISA):
//   A 16x4 : lanes 0-15 -> row=lane, K+{0,1}; lanes 16-31 -> K+{2,3}
//   B 4x16 : col=lane%16; lanes 0-15 -> K+{0,1}; lanes 16-31 -> K+{2,3}
//   C 16x16: VGPR i -> row i (lanes 0-15) / row 8+i (lanes 16-31), col=lane%16
// ---------------------------------------------------------------------------
__global__ __launch_bounds__(256) void gcn_gemm_wmma(
    const float* __restrict__ A, const float* __restrict__ B,
    const float* __restrict__ dinv, const float* __restrict__ bias,
    float* __restrict__ m, float* __restrict__ agg, int N) {
  __shared__ float As[32 * HCH];  // 16 KB
  const int tid = threadIdx.x;
  const int tileM = blockIdx.x;   // 32-row tile

  // stage A tile: 4096 floats = 1024 float4, 4 per thread
  {
    const float4* src = (const float4*)A;
    float4* dst = (float4*)As;
    const int base4 = tileM * 1024;  // float4 index of tile start
    const int lim4 = N * (HCH / 4);
#pragma unroll
    for (int j = 0; j < 4; ++j) {
      int i = base4 + tid + j * 256;
      dst[tid + j * 256] = (i < lim4) ? src[i] : make_float4(0.f, 0.f, 0.f, 0.f);
    }
  }
  __syncthreads();

  const int wave = tid >> 5;          // col tile 0..7
  const int lane = tid & 31;
  const int col = wave * 16 + (lane & 15);
  const int lrow = lane & 15;         // A row within 16-row subtile
  const int khalf = (lane >> 4) * 2;  // lanes 0-15: K+0,1 ; lanes 16-31: K+2,3

  v8f acc0 = {};  // rows tileM*32 + 0..15
  v8f acc1 = {};  // rows tileM*32 + 16..31
#pragma unroll
  for (int kc = 0; kc < 32; ++kc) {
    const int kb = kc * 4 + khalf;
    v2f b, a0, a1;
    b.x = B[kb * HCH + col];
    b.y = B[(kb + 1) * HCH + col];
    a0.x = As[lrow * HCH + kb];
    a0.y = As[lrow * HCH + kb + 1];
    a1.x = As[(16 + lrow) * HCH + kb];
    a1.y = As[(16 + lrow) * HCH + kb + 1];
    acc0 = __builtin_amdgcn_wmma_f32_16x16x4_f32(
        false, a0, false, b, (short)0, acc0, false, false);
    acc1 = __builtin_amdgcn_wmma_f32_16x16x4_f32(
        false, a1, false, b, (short)0, acc1, false, false);
  }

  // fused epilogue: write m and agg = bias + m*dinv^2
  const float bc = bias[col];
  const int rbase = tileM * 32 + (lane >> 4) * 8;
#pragma unroll
  for (int i = 0; i < 8; ++i) {
    int r0 = rbase + i;
    int r1 = r0 + 16;
    if (r0 < N) {
      float di = dinv[r0];
      size_t off = (size_t)r0 * HCH + col;
      m[off] = acc0[i];
      agg[off] = fmaf(acc0[i], di * di, bc);
    }
    if (r1 < N) {
      float di = dinv[r1];
      size_t off = (size_t)r1 * HCH + col;
      m[off] = acc1[i];
      agg[off] = fmaf(acc1[i], di * di, bc);
    }
  }
}

// ---------------------------------------------------------------------------
// Edge scatter: one wave per edge; lane handles 4 channels (float4 slice).
// agg[dst] += m[src] * (dinv[src]*dinv[dst])
// ---------------------------------------------------------------------------
__global__ __launch_bounds__(256) void gcn_scatter_edges(
    const float* __restrict__ m, const float* __restrict__ dinv,
    const int* __restrict__ ei, float* __restrict__ agg, int E) {
  int wid = (blockIdx.x * blockDim.x + threadIdx.x) >> 5;
  if (wid >= E) return;
  int lane = threadIdx.x & 31;
  int s = ei[wid];
  int d = ei[E + wid];
  float nrm = dinv[s] * dinv[d];
  float4 v = ((const float4*)(m + (size_t)s * HCH))[lane];
  float* ap = agg + (size_t)d * HCH + lane * 4;
  atomicAdd(ap + 0, v.x * nrm);
  atomicAdd(ap + 1, v.y * nrm);
  atomicAdd(ap + 2, v.z * nrm);
  atomicAdd(ap + 3, v.w * nrm);
}

// ---------------------------------------------------------------------------
// BN stats: per-channel sum / sumsq of relu(agg), coalesced row sweeps,
// atomics into stats[0:128]=sum, stats[128:256]=sumsq
// ---------------------------------------------------------------------------
__global__ __launch_bounds__(HCH) void gcn_bn_partial(
    const float* __restrict__ agg, float* __restrict__ stats, int N) {
  int c = threadIdx.x;
  float s = 0.f, sq = 0.f;
  for (int n = blockIdx.x; n < N; n += gridDim.x) {
    float a = fmaxf(agg[(size_t)n * HCH + c], 0.f);
    s += a;
    sq = fmaf(a, a, sq);
  }
  atomicAdd(&stats[c], s);
  atomicAdd(&stats[HCH + c], sq);
}

__global__ __launch_bounds__(HCH) void gcn_bn_finalize(float* __restrict__ stats, int N) {
  int c = threadIdx.x;
  float inv = 1.0f / (float)N;
  float mean = stats[c] * inv;
  float var = stats[HCH + c] * inv - mean * mean;
  stats[c] = mean;
  stats[HCH + c] = var;
}

__global__ __launch_bounds__(256) void gcn_bn_apply(
    const float* __restrict__ agg, const float* __restrict__ stats,
    const float* __restrict__ gamma, const float* __restrict__ beta,
    float* __restrict__ h, int total) {
  int i = blockIdx.x * blockDim.x + threadIdx.x;
  if (i >= total) return;
  int c = i & (HCH - 1);
  float a = fmaxf(agg[i], 0.f);
  float bn = fmaf(gamma[c] * rsqrtf(stats[HCH + c] + 1e-5f), a - stats[c], beta[c]);
  h[i] = fmaxf(bn, 0.f) + h[i];   // relu + residual, in place
}

// ---------------------------------------------------------------------------
// Pooling: wave per node, lane per 4 channels; counts via lane 0
// ---------------------------------------------------------------------------
__global__ __launch_bounds__(256) void gcn_pool_scatter(
    const float* __restrict__ h, const int* __restrict__ batch,
    float* __restrict__ pooled, float* __restrict__ counts, int N) {
  int wid = (blockIdx.x * blockDim.x + threadIdx.x) >> 5;
  if (wid >= N) return;
  int lane = threadIdx.x & 31;
  int g = batch[wid];
  float4 v = ((const float4*)(h + (size_t)wid * HCH))[lane];
  float* pp = pooled + (size_t)g * HCH + lane * 4;
  atomicAdd(pp + 0, v.x);
  atomicAdd(pp + 1, v.y);
  atomicAdd(pp + 2, v.z);
  atomicAdd(pp + 3, v.w);
  if (lane == 0) atomicAdd(&counts[g], 1.0f);
}

// ---------------------------------------------------------------------------
// Readout: wave per graph: sigmoid(dot(pooled/cnt, lin_W) + lin_b)
// ---------------------------------------------------------------------------
__global__ __launch_bounds__(256) void gcn_readout(
    const float* __restrict__ pooled, const float* __restrict__ counts,
    const float* __restrict__ lin_W, const float* __restrict__ lin_b,
    float* __restrict__ out, int G) {
  int wid = (blockIdx.x * blockDim.x + threadIdx.x) >> 5;
  if (wid >= G) return;
  int lane = threadIdx.x & 31;
  float acc = 0.f;
#pragma unroll
  for (int i = 0; i < 4; ++i) {
    int c = lane + 32 * i;
    acc = fmaf(pooled[(size_t)wid * HCH + c], lin_W[c], acc);
  }
#pragma unroll
  for (int off = 16; off > 0; off >>= 1) acc += __shfl_xor(acc, off, 32);
  if (lane == 0) {
    float cnt = fmaxf(counts[wid], 1.0f);
    float z = acc / cnt + lin_b[0];
    out[wid] = 1.0f / (1.0f + __expf(-z));
  }
}

// ---------------------------------------------------------------------------
extern "C" void kernel_launch(void* const* d_in, const int* in_sizes, int n_in,
                              void* d_out, int out_size, void* d_ws, size_t ws_size,
                              hipStream_t stream) {
  const int* x          = (const int*)d_in[0];     // [N,9]
  const int* edge_index = (const int*)d_in[1];     // [2,E]
  const int* batch_idx  = (const int*)d_in[2];     // [N]
  const float* atom_emb = (const float*)d_in[3];   // [9,119,128]
  const float* conv_W   = (const float*)d_in[4];   // [4,128,128]
  const float* conv_b   = (const float*)d_in[5];   // [4,128]
  const float* bn_gamma = (const float*)d_in[6];   // [4,128]
  const float* bn_beta  = (const float*)d_in[7];   // [4,128]
  const float* lin_W    = (const float*)d_in[8];   // [128,1]
  const float* lin_b    = (const float*)d_in[9];   // [1]
  float* out = (float*)d_out;                      // [G,1]

  const int N = in_sizes[2];
  const int E = in_sizes[1] / 2;
  const int G = out_size;
  const int L = 4;
  const size_t NH = (size_t)N * HCH;

  // workspace carve
  float* h      = (float*)d_ws;
  float* m      = h + NH;
  float* agg    = m + NH;
  float* dinv   = agg + NH;
  float* stats  = dinv + N;            // [2*128] sum/sumsq -> mean/var
  float* pooled = stats + 2 * HCH;     // [G,128]
  float* counts = pooled + (size_t)G * HCH;  // [G]

  const int T = 256;
  const int total = (int)NH;

  // 1) atom encoder
  gcn_atom_encode<<<N, HCH, 0, stream>>>(x, atom_emb, h, N);

  // 2) degrees -> dinv  (init 1.0 for self-loop, add in-degree, rsqrt)
  gcn_fill<<<(N + T - 1) / T, T, 0, stream>>>(dinv, 1.0f, N);
  gcn_count_deg<<<(E + T - 1) / T, T, 0, stream>>>(edge_index, dinv, E);
  gcn_rsqrt_inplace<<<(N + T - 1) / T, T, 0, stream>>>(dinv, N);

  // 3) layers
  const int mTiles = (N + 31) / 32;
  const int edgeBlocks = (int)(((long long)E * 32 + T - 1) / T);
  for (int l = 0; l < L; ++l) {
    gcn_gemm_wmma<<<mTiles, 256, 0, stream>>>(
        h, conv_W + (size_t)l * HCH * HCH, dinv, conv_b + l * HCH, m, agg, N);
    gcn_scatter_edges<<<edgeBlocks, T, 0, stream>>>(m, dinv, edge_index, agg, E);
    gcn_fill<<<1, 2 * HCH, 0, stream>>>(stats, 0.0f, 2 * HCH);
    gcn_bn_partial<<<512, HCH, 0, stream>>>(agg, stats, N);
    gcn_bn_finalize<<<1, HCH, 0, stream>>>(stats, N);
    gcn_bn_apply<<<(total + T - 1) / T, T, 0, stream>>>(
        agg, stats, bn_gamma + l * HCH, bn_beta + l * HCH, h, total);
  }

  // 4) readout
  int gh = G * HCH;
  gcn_fill<<<(gh + T - 1) / T, T, 0, stream>>>(pooled, 0.0f, gh);
  gcn_fill<<<(G + T - 1) / T, T, 0, stream>>>(counts, 0.0f, G);
  gcn_pool_scatter<<<(int)(((long long)N * 32 + T - 1) / T), T, 0, stream>>>(
      h, batch_idx, pooled, counts, N);
  gcn_readout<<<(int)(((long long)G * 32 + T - 1) / T), T, 0, stream>>>(
      pooled, counts, lin_W, lin_b, out, G);
}